// Model_39676907887665
// MI455X (gfx1250) — compile-verified
//
#include <hip/hip_runtime.h>
#include <math.h>

typedef float v2f __attribute__((ext_vector_type(2)));
typedef float v8f __attribute__((ext_vector_type(8)));

#define SQ 2048
#define SK 2048
#define DD 128    // head dim (contraction for Q.K^T)
#define DV 128    // value dim
#define CHUNK 128 // Sk rows per block in the K^T.V kernel

// Issue one per-lane 16B async global->LDS copy (ASYNCcnt-tracked, no VGPR data).
__device__ __forceinline__ void async_copy_b128(unsigned lds_addr, unsigned long long gaddr) {
    asm volatile("global_load_async_to_lds_b128 %0, %1, off"
                 :: "v"(lds_addr), "v"(gaddr) : "memory");
}
__device__ __forceinline__ void wait_async_all() {
    asm volatile("s_wait_asynccnt 0x0" ::: "memory");
}

// ---------------------------------------------------------------------------
// Kernel 0: zero the M = K^T V accumulator scratch (atomic accumulation target)
// ---------------------------------------------------------------------------
__global__ void fa_zero_kernel(float* __restrict__ p, int n) {
    int i = blockIdx.x * blockDim.x + threadIdx.x;
    if (i < n) p[i] = 0.0f;
}

// ---------------------------------------------------------------------------
// Kernel 1: M[b] += s * K[b]^T V[b] over one 128-row chunk of Sk.
//   grid = (SK/CHUNK, B), block = 256 threads (8 waves, wave32).
//   V chunk (64 KB) is staged in LDS via async loads; all 8 waves share it.
//   Wave w owns rows [16w,16w+16) of M (8 col-tiles of 16x16).
// ---------------------------------------------------------------------------
__global__ __launch_bounds__(256) void fa_ktv_kernel(
    const float* __restrict__ K, const float* __restrict__ V,
    const float* __restrict__ qk_scale, float* __restrict__ M) {
    __shared__ float sV[CHUNK * DV];      // 64 KB of the 320 KB WGP pool

    const int b     = blockIdx.y;
    const int kbase = blockIdx.x * CHUNK;
    const int tid   = threadIdx.x;
    const int wave  = tid >> 5;           // tile-row of M
    const int lane  = tid & 31;
    const int half  = lane >> 4;          // 0: K=k0,k0+1   1: K=k0+2,k0+3
    const int l16   = lane & 15;

    const float* __restrict__ Kb = K + (size_t)b * SK * DD;
    const float* __restrict__ Vb = V + (size_t)b * SK * DV;
    float* __restrict__ Mb       = M + (size_t)b * DD * DV;

    // --- async stage: V[kbase:kbase+CHUNK, :] -> LDS (linear copy, b128/lane)
    {
        const unsigned lbase = (unsigned)(uintptr_t)sV;            // LDS offset
        const unsigned long long gbase =
            (unsigned long long)(uintptr_t)(Vb + (size_t)kbase * DV);
        #pragma unroll
        for (int it = 0; it < (CHUNK * DV * 4) / (256 * 16); ++it) { // 16 iters
            const unsigned off = (unsigned)(it * 256 + tid) * 16u;
            async_copy_b128(lbase + off, gbase + off);
        }
        wait_async_all();
        __syncthreads();
    }

    const int m = 16 * wave + l16;        // row of M == column of K

    v8f acc[8] = {};                      // 8 col-tiles, 16x16 f32 each

    for (int kk = 0; kk < CHUNK / 4; ++kk) {
        const int kl = kk * 4 + half * 2;           // chunk-local K index
        const int kg = kbase + kl;                  // global Sk index
        v2f a;                                      // A fragment: K^T
        a.x = Kb[(size_t)kg       * DD + m];
        a.y = Kb[(size_t)(kg + 1) * DD + m];
        #pragma unroll
        for (int nt = 0; nt < 8; ++nt) {
            const int n = 16 * nt + l16;
            v2f bf;                                 // B fragment: V from LDS
            bf.x = sV[kl       * DV + n];
            bf.y = sV[(kl + 1) * DV + n];
            acc[nt] = __builtin_amdgcn_wmma_f32_16x16x4_f32(
                false, a, false, bf, (short)0, acc[nt], false, false);
        }
    }

    // scale once here (linear), then accumulate partial chunks across blocks
    const float s = 1.0f / (sqrtf((float)DD) * qk_scale[0]);
    #pragma unroll
    for (int nt = 0; nt < 8; ++nt) {
        #pragma unroll
        for (int r = 0; r < 8; ++r) {
            const int row = 16 * wave + r + 8 * half;   // C/D VGPR layout
            const int col = 16 * nt + l16;
            atomicAdd(&Mb[row * DV + col], acc[nt][r] * s);
        }
    }
}

// ---------------------------------------------------------------------------
// Kernel 2: O[b] = Q[b] * M[b].
//   grid = (SQ/64, B), block = 128 threads (4 waves); the whole 64 KB M[b]
//   is staged in LDS via async loads. Wave w owns 16 Q-rows -> 16x128 strip.
// ---------------------------------------------------------------------------
__global__ __launch_bounds__(128) void fa_qm_kernel(
    const float* __restrict__ Q, const float* __restrict__ M,
    float* __restrict__ O) {
    __shared__ float sM[DD * DV];         // 64 KB

    const int b    = blockIdx.y;
    const int tid  = threadIdx.x;
    const int wave = tid >> 5;
    const int lane = tid & 31;
    const int half = lane >> 4;
    const int l16  = lane & 15;
    const int r0   = blockIdx.x * 64 + wave * 16;

    const float* __restrict__ Qb = Q + (size_t)b * SQ * DD;
    const float* __restrict__ Mb = M + (size_t)b * DD * DV;
    float* __restrict__ Ob       = O + (size_t)b * SQ * DV;

    // --- async stage: M[b] -> LDS (linear copy, b128/lane, 32 iterations)
    {
        const unsigned lbase = (unsigned)(uintptr_t)sM;
        const unsigned long long gbase = (unsigned long long)(uintptr_t)Mb;
        #pragma unroll
        for (int it = 0; it < (DD * DV * 4) / (128 * 16); ++it) {   // 32 iters
            const unsigned off = (unsigned)(it * 128 + tid) * 16u;
            async_copy_b128(lbase + off, gbase + off);
        }
        wait_async_all();
        __syncthreads();
    }

    const int m = r0 + l16;               // this lane's Q row

    v8f acc[8] = {};

    for (int kk = 0; kk < DD / 4; ++kk) {
        const int k0 = kk * 4 + half * 2;
        v2f a;                            // 2 consecutive floats -> b64 load
        a.x = Qb[(size_t)m * DD + k0];
        a.y = Qb[(size_t)m * DD + k0 + 1];
        #pragma unroll
        for (int nt = 0; nt < 8; ++nt) {
            const int n = 16 * nt + l16;
            v2f bf;                       // B fragment: M from LDS
            bf.x = sM[k0       * DV + n];
            bf.y = sM[(k0 + 1) * DV + n];
            acc[nt] = __builtin_amdgcn_wmma_f32_16x16x4_f32(
                false, a, false, bf, (short)0, acc[nt], false, false);
        }
    }

    #pragma unroll
    for (int nt = 0; nt < 8; ++nt) {
        #pragma unroll
        for (int r = 0; r < 8; ++r) {
            const int row = r0 + r + 8 * half;
            const int col = 16 * nt + l16;
            Ob[(size_t)row * DV + col] = acc[nt][r];
        }
    }
}

// ---------------------------------------------------------------------------
// Inputs (setup_inputs order): 0=qk (dummy), 1=qk_scaling_factor [1],
// 2=query [B,Sq,D], 3=key [B,Sk,D], 4=value [B,Sk,Dv].  Output: [B,Sq,Dv] f32.
// ---------------------------------------------------------------------------
extern "C" void kernel_launch(void* const* d_in, const int* in_sizes, int n_in,
                              void* d_out, int out_size, void* d_ws, size_t ws_size,
                              hipStream_t stream) {
    (void)n_in; (void)out_size; (void)ws_size;
    const float* qk_scale = (const float*)d_in[1];
    const float* Q        = (const float*)d_in[2];
    const float* K        = (const float*)d_in[3];
    const float* V        = (const float*)d_in[4];
    float* O = (float*)d_out;
    float* M = (float*)d_ws;                      // [B,128,128] accumulator

    const int B  = in_sizes[2] / (SQ * DD);       // = 8 for this problem
    const int nM = B * DD * DV;

    fa_zero_kernel<<<dim3((nM + 255) / 256), 256, 0, stream>>>(M, nM);
    fa_ktv_kernel<<<dim3(SK / CHUNK, B), 256, 0, stream>>>(K, V, qk_scale, M);
    fa_qm_kernel <<<dim3(SQ / 64,  B), 128, 0, stream>>>(Q, M, O);
}